// GS_DMTET_49314814493119
// MI455X (gfx1250) — compile-verified
//
#include <hip/hip_runtime.h>
#include <hip/hip_bf16.h>
#include <stdint.h>

// ---------------------------------------------------------------------------
// Marching-tets (DMTet) structure + differentiable edge interpolation.
// Bandwidth/atomic-bound workload: no matmul -> no WMMA. CDNA5 paths used:
//   * TDM tensor_load_to_lds to stage tet indices into LDS (TENSORcnt)
//   * s_wait_tensorcnt
//   * wave32 ballot-aggregated atomic id allocation
//   * global_prefetch_b8
//   * ds_load_b128 reads of the staged tile
// ---------------------------------------------------------------------------

typedef unsigned int v4u __attribute__((ext_vector_type(4)));
typedef int          v8i __attribute__((ext_vector_type(8)));
typedef int          v4i __attribute__((ext_vector_type(4)));

#define EMPTY_KEY 0xFFFFFFFFFFFFFFFFULL

__constant__ int c_tri[16][6] = {
    {-1,-1,-1,-1,-1,-1},{1,0,2,-1,-1,-1},{4,0,3,-1,-1,-1},{1,4,2,1,3,4},
    {3,1,5,-1,-1,-1},{2,3,0,2,5,3},{1,4,0,1,5,4},{4,2,5,-1,-1,-1},
    {4,5,2,-1,-1,-1},{4,1,0,4,5,1},{3,2,0,3,5,2},{1,3,5,-1,-1,-1},
    {4,1,2,4,3,1},{3,0,4,-1,-1,-1},{2,0,1,-1,-1,-1},{-1,-1,-1,-1,-1,-1}};
__constant__ int c_ntri[16] = {0,1,1,2,1,2,2,1,1,2,2,1,2,1,1,0};
__constant__ int c_ea[6] = {0,0,0,1,1,2};
__constant__ int c_eb[6] = {1,2,3,2,3,3};

__device__ __forceinline__ unsigned hash64(unsigned long long x) {
    x ^= x >> 33; x *= 0xff51afd7ed558ccdULL;
    x ^= x >> 33; x *= 0xc4ceb9fe1a85ec53ULL;
    x ^= x >> 33;
    return (unsigned)x;
}

// Wave32 aggregated counter allocation: one atomic per wave, not per lane.
__device__ __forceinline__ int warp_alloc(int* counter, bool want) {
    unsigned mask = (unsigned)__ballot(want ? 1 : 0);
    int lane = (int)(threadIdx.x & 31);
    int base = 0;
    if (mask) {
        int leader = __ffs(mask) - 1;
        if (lane == leader) base = atomicAdd(counter, __popc(mask));
        base = __shfl(base, leader, 32);
    }
    return base + __popc(mask & ((1u << lane) - 1u));
}

__global__ void init_kernel(unsigned long long* __restrict__ keys,
                            int* __restrict__ counters, unsigned cap) {
    unsigned i = blockIdx.x * 256u + threadIdx.x;
    if (i < cap) keys[i] = EMPTY_KEY;
    if (i < 8u) counters[i] = 0;
}

// Pass 1: classify tets, hash-insert crossing edges, compute interpolated verts.
__global__ void __launch_bounds__(256)
edge_kernel(const int* __restrict__ tets, const float* __restrict__ sdf,
            const float* __restrict__ verts, const float* __restrict__ deform,
            const int* __restrict__ gsz, int n,
            unsigned long long* __restrict__ keys, int* __restrict__ vals,
            int* __restrict__ counters, float* __restrict__ out,
            long long out_cap, unsigned hmask) {
    __shared__ __align__(16) int lds[1024];   // 256 tets * 4 int32

    // --- TDM: each wave DMAs its 32 tets (512 B) into its LDS slice ---
    const int waveU = __builtin_amdgcn_readfirstlane((int)(threadIdx.x >> 5));
    long long startDw = ((long long)blockIdx.x * 256 + (long long)waveU * 32) * 4;
    long long totalDw = (long long)n * 4;
    long long rem = totalDw - startDw;
    int dwords = (rem >= 128) ? 128 : (rem > 0 ? (int)rem : 0);
    if (dwords > 0) {
        unsigned long long gaddr =
            (unsigned long long)(uintptr_t)tets + (unsigned long long)startDw * 4u;
        unsigned laddr = (unsigned)(uintptr_t)(&lds[waveU * 128]);
        unsigned ud = (unsigned)dwords;
        v4u g0;
        g0[0] = 1u;                                        // count=1 descriptor
        g0[1] = laddr;                                     // lds_addr
        g0[2] = (unsigned)(gaddr & 0xFFFFFFFFull);         // global_addr lo
        g0[3] = (unsigned)((gaddr >> 32) & 0x01FFFFFFull)  // global_addr hi
                | (2u << 30);                              // type=2 (image)
        v8i g1;
        g1[0] = (int)(2u << 16);                 // data_size = 4B, no mask
        g1[1] = (int)(ud << 16);                 // tensor_dim0[15:0]
        g1[2] = (int)((ud >> 16) | (1u << 16));  // tensor_dim0[31:16], tensor_dim1=1
        g1[3] = (int)(ud << 16);                 // tile_dim0 = dwords
        g1[4] = 0;                               // tile_dim1=0, tile_dim2=0
        g1[5] = (int)ud;                         // tensor_dim0_stride
        g1[6] = 0;
        g1[7] = 0;
        v4i gz4 = {0, 0, 0, 0};
        v8i gz8 = {0, 0, 0, 0, 0, 0, 0, 0};
        __builtin_amdgcn_tensor_load_to_lds(g0, g1, gz4, gz4, gz8, 0);
    }
    __builtin_amdgcn_s_wait_tensorcnt(0);

    int t = blockIdx.x * 256 + threadIdx.x;
    if (t >= n) return;

    const int4 tv = *(const int4*)(&lds[threadIdx.x * 4]);  // ds_load_b128
    int vv[4] = {tv.x, tv.y, tv.z, tv.w};
    float ss[4];
    int oc[4];
    int sum = 0;
#pragma unroll
    for (int k = 0; k < 4; ++k) {
        ss[k] = sdf[vv[k]];
        oc[k] = (ss[k] > 0.0f) ? 1 : 0;
        sum += oc[k];
    }
    if (sum == 0 || sum == 4) return;   // not a surface-crossing tet

    const float invg = 1.0f / (float)gsz[0];

#pragma unroll
    for (int e = 0; e < 6; ++e) {
        int ia = c_ea[e], ib = c_eb[e];
        int a = vv[ia], b = vv[ib];
        bool crossing = (oc[ia] != oc[ib]);
        bool won = false;
        unsigned slot = 0;
        int lo = a < b ? a : b;
        int hi = a < b ? b : a;
        if (crossing) {
            unsigned long long key =
                ((unsigned long long)(unsigned)lo << 32) | (unsigned)hi;
            unsigned h = hash64(key) & hmask;
            while (true) {
                unsigned long long old = atomicCAS(&keys[h], EMPTY_KEY, key);
                if (old == EMPTY_KEY) { won = true; slot = h; break; }
                if (old == key) break;     // someone else owns this edge
                h = (h + 1u) & hmask;
            }
        }
        int id = warp_alloc(&counters[0], won);   // wave-aggregated alloc
        if (won) {
            vals[slot] = id;
            float s0 = sdf[lo], s1 = sdf[hi];
            float inv = 1.0f / (s0 - s1);
            float w0 = -s1 * inv, w1 = s0 * inv;
            long long base = 3LL * (long long)id;
            if (base + 2 < out_cap) {
#pragma unroll
                for (int k = 0; k < 3; ++k) {
                    float p0 = verts[3LL * lo + k] + tanhf(deform[3LL * lo + k]) * invg;
                    float p1 = verts[3LL * hi + k] + tanhf(deform[3LL * hi + k]) * invg;
                    out[base + k] = p0 * w0 + p1 * w1;
                }
            }
        }
    }
}

__device__ __forceinline__ int hash_find(const unsigned long long* __restrict__ keys,
                                         const int* __restrict__ vals,
                                         unsigned long long key, unsigned hmask) {
    unsigned h = hash64(key) & hmask;
    for (unsigned it = 0; it <= hmask; ++it) {
        unsigned long long k = keys[h];
        if (k == key) return vals[h];
        if (k == EMPTY_KEY) return -1;
        h = (h + 1u) & hmask;
    }
    return -1;
}

// Pass 2: emit triangles (edge ids) after the vertex block (offset 3*E).
__global__ void __launch_bounds__(256)
face_kernel(const int* __restrict__ tets, const float* __restrict__ sdf, int n,
            const unsigned long long* __restrict__ keys,
            const int* __restrict__ vals, int* __restrict__ counters,
            float* __restrict__ out, long long out_cap, unsigned hmask) {
    int t = blockIdx.x * 256 + threadIdx.x;
    // Pull upcoming tet cachelines toward L2 (global_prefetch_b8).
    long long pf = 4LL * (t + 8LL * 256);
    if (pf < 4LL * n) __builtin_prefetch((const void*)(tets + pf), 0, 1);

    int4 tv = {0, 0, 0, 0};
    bool inr = (t < n);
    if (inr) tv = *(const int4*)(tets + 4LL * t);
    int vv[4] = {tv.x, tv.y, tv.z, tv.w};
    int cfg = 0, sum = 0;
#pragma unroll
    for (int k = 0; k < 4; ++k) {
        int o = inr && (sdf[vv[k]] > 0.0f) ? 1 : 0;
        cfg |= o << k;
        sum += o;
    }
    int c = (inr && sum > 0 && sum < 4) ? c_ntri[cfg] : 0;

    // Wave32 aggregated allocation of (0|1|2) triangle slots per lane.
    unsigned b1 = (unsigned)__ballot(c >= 1);
    unsigned b2 = (unsigned)__ballot(c >= 2);
    int lane = (int)(threadIdx.x & 31);
    unsigned lt = (1u << lane) - 1u;
    int total = __popc(b1) + __popc(b2);
    int base = 0;
    if (total) {
        int leader = __ffs(b1 | b2) - 1;
        if (lane == leader) base = atomicAdd(&counters[1], total);
        base = __shfl(base, leader, 32);
    }
    if (c == 0) return;

    int E = counters[0];          // final edge count from pass 1 (stream order)
    long long fb = 3LL * (long long)E;

    for (int tri = 0; tri < c; ++tri) {
        int slot = (tri == 0) ? (base + __popc(b1 & lt))
                              : (base + __popc(b1) + __popc(b2 & lt));
#pragma unroll
        for (int k = 0; k < 3; ++k) {
            int es = c_tri[cfg][tri * 3 + k];
            int a = vv[c_ea[es]], b = vv[c_eb[es]];
            int lo = a < b ? a : b;
            int hi = a < b ? b : a;
            unsigned long long key =
                ((unsigned long long)(unsigned)lo << 32) | (unsigned)hi;
            int id = hash_find(keys, vals, key, hmask);
            long long idx = fb + 3LL * slot + k;
            if (idx >= 0 && idx < out_cap) out[idx] = (float)id;
        }
    }
}

extern "C" void kernel_launch(void* const* d_in, const int* in_sizes, int n_in,
                              void* d_out, int out_size, void* d_ws, size_t ws_size,
                              hipStream_t stream) {
    const float* verts  = (const float*)d_in[0];
    const float* deform = (const float*)d_in[1];
    const float* sdf    = (const float*)d_in[2];
    const int*   tets   = (const int*)d_in[3];
    const int*   gsz    = (const int*)d_in[4];
    (void)n_in;

    int n = in_sizes[3] / 4;
    float* out = (float*)d_out;

    // Size the hash table from the available workspace (power of two).
    unsigned cap = 1u << 12;
    while (cap < (1u << 23) &&
           ((size_t)(cap << 1) * 12u + 64u) <= ws_size) cap <<= 1;
    unsigned hmask = cap - 1u;

    int* counters = (int*)d_ws;
    unsigned long long* keys = (unsigned long long*)((char*)d_ws + 64);
    int* vals = (int*)(keys + cap);

    init_kernel<<<(cap + 255u) / 256u, 256, 0, stream>>>(keys, counters, cap);
    if (n > 0) {
        int blocks = (n + 255) / 256;
        edge_kernel<<<blocks, 256, 0, stream>>>(tets, sdf, verts, deform, gsz, n,
                                                keys, vals, counters, out,
                                                (long long)out_size, hmask);
        face_kernel<<<blocks, 256, 0, stream>>>(tets, sdf, n, keys, vals, counters,
                                                out, (long long)out_size, hmask);
    }
}